// TorchRNN_12249246728314
// MI455X (gfx1250) — compile-verified
//
#include <hip/hip_runtime.h>
#include <stdint.h>

// Problem constants (from reference): B=16384, T=128, I=26, H=4, C=5
#define B_TOT   16384
#define T_LEN   128
#define I_DIM   26
#define H_DIM   4
#define C_DIM   5

// Tiling: each wave (32 lanes) owns 32 batch rows; time processed in chunks.
#define TCHUNK      8
#define NCHUNK      (T_LEN / TCHUNK)            // 16 chunks
#define ROW_DW      (TCHUNK * I_DIM)            // 208 dwords per row-chunk
#define ROW_DW_PAD  (ROW_DW + (ROW_DW >> 4))    // 221 (TDM pads 1 dw per 16 dw)
#define BUF_DW      (32 * ROW_DW_PAD)           // 7072 dwords per wave buffer
#define BUF_BYTES   (BUF_DW * 4)                // 28288 B
#define WAVES_PB    2
#define BLOCK_T     (WAVES_PB * 32)             // 64 threads
#define LDS_BYTES   (WAVES_PB * 2 * BUF_BYTES)  // double buffered: 113152 B

#if __has_builtin(__builtin_amdgcn_tensor_load_to_lds) && __has_builtin(__builtin_amdgcn_s_wait_tensorcnt)
#define HAS_TDM 1
#else
#define HAS_TDM 0
#endif

#if __has_builtin(__builtin_amdgcn_wmma_f32_16x16x4_f32)
#define HAS_WMMA4 1
#else
#define HAS_WMMA4 0
#endif

#if __has_builtin(__builtin_amdgcn_tanhf)
#define TANHF(v) __builtin_amdgcn_tanhf(v)
#else
#define TANHF(v) tanhf(v)
#endif

typedef __attribute__((ext_vector_type(4))) unsigned int v4u;
typedef __attribute__((ext_vector_type(8))) int   v8i_;
typedef __attribute__((ext_vector_type(4))) int   v4i_;
typedef __attribute__((ext_vector_type(8))) float v8f_;
typedef __attribute__((ext_vector_type(2))) float v2f_;

typedef __attribute__((address_space(3))) void lds_void_t;

__device__ __forceinline__ unsigned lds_byte_offset(void* p) {
  // generic -> LDS addrspace -> 32-bit LDS byte address (what the D# wants)
  return (unsigned)(uintptr_t)(lds_void_t*)p;
}

#if HAS_TDM
// One TDM descriptor: 3D tile  dim0=26 floats (contig), dim1=TCHUNK (stride I),
// dim2=32 rows (stride T*I).  LDS padding: 1 dword after every 16 dwords
// (pad_enable=1, pad_interval code 3, pad_amount code 0) -> conflict-free reads.
__device__ __forceinline__ void tdm_load_tile(const float* gsrc, unsigned lds_addr) {
  unsigned long long ga = (unsigned long long)(uintptr_t)gsrc;
  v4u g0 = {
    1u,                                                   // count=1, user mode, gather off
    lds_addr,                                             // lds_addr (bytes)
    (unsigned)(ga & 0xFFFFFFFFu),                         // global_addr[31:0]
    (unsigned)((ga >> 32) & 0x01FFFFFFull) | 0x80000000u  // ga[56:32] | type=2
  };
  v8i_ g1 = {
    (int)((2u << 16) | (1u << 20) | (3u << 22)),          // data_size=4B, pad 1dw/16dw
    (int)(((unsigned)I_DIM) << 16),                       // tensor_dim0 lo16
    (int)(((unsigned)TCHUNK) << 16),                      // dim0 hi16=0 | tensor_dim1 lo16
    (int)(((unsigned)I_DIM) << 16),                       // dim1 hi16=0 | tile_dim0=26
    (int)(((unsigned)TCHUNK) | (32u << 16)),              // tile_dim1=8 | tile_dim2=32
    (int)I_DIM,                                           // tensor_dim0_stride lo32 = 26
    (int)(((unsigned)(T_LEN * I_DIM)) << 16),             // d0s hi16=0 | d1 stride lo16 = 3328
    (int)(((unsigned)(T_LEN * I_DIM)) >> 16)              // d1 stride [47:16]
  };
  v4i_ g2 = { 32, 0, 0, 0 };                              // tensor_dim2=32; dim3 unused
  v4i_ g3 = { 0, 0, 0, 0 };
  v8i_ g4 = { 0, 0, 0, 0, 0, 0, 0, 0 };                   // extension/gather group (unused)
  __builtin_amdgcn_tensor_load_to_lds(g0, g1, g2, g3, g4, 0);
}
#endif

__global__ __launch_bounds__(BLOCK_T) void rnn_fused(
    const float* __restrict__ x,    const float* __restrict__ W_ih,
    const float* __restrict__ W_hh, const float* __restrict__ W_fc,
    const float* __restrict__ b_fc, float* __restrict__ out) {
  extern __shared__ float smem[];
  const int lane    = threadIdx.x & 31;
  const int wave    = threadIdx.x >> 5;
  const int rowBase = blockIdx.x * BLOCK_T + wave * 32;   // first batch row of this wave

  float* buf0 = smem + wave * 2 * BUF_DW;
  float* buf1 = buf0 + BUF_DW;

  // W_hh into registers (uniform -> scalar loads)
  float whh[H_DIM][H_DIM];
#pragma unroll
  for (int a = 0; a < H_DIM; ++a)
#pragma unroll
    for (int b = 0; b < H_DIM; ++b) whh[a][b] = W_hh[a * H_DIM + b];

  float h0 = 0.f, h1 = 0.f, h2 = 0.f, h3 = 0.f;

  auto compute_chunk = [&](const float* cbuf) {
    const float* r = cbuf + lane * ROW_DW_PAD;
#pragma unroll 2
    for (int tt = 0; tt < TCHUNK; ++tt) {
      float a0 = 0.f, a1 = 0.f, a2 = 0.f, a3 = 0.f;
      const int mb = tt * I_DIM;
#pragma unroll
      for (int i = 0; i < I_DIM; ++i) {
        const int m = mb + i;
        const float xv = r[m + (m >> 4)];                 // padded LDS mapping
        a0 = fmaf(W_ih[i],              xv, a0);
        a1 = fmaf(W_ih[I_DIM + i],      xv, a1);
        a2 = fmaf(W_ih[2 * I_DIM + i],  xv, a2);
        a3 = fmaf(W_ih[3 * I_DIM + i],  xv, a3);
      }
      const float z0 = a0 + whh[0][0]*h0 + whh[0][1]*h1 + whh[0][2]*h2 + whh[0][3]*h3;
      const float z1 = a1 + whh[1][0]*h0 + whh[1][1]*h1 + whh[1][2]*h2 + whh[1][3]*h3;
      const float z2 = a2 + whh[2][0]*h0 + whh[2][1]*h1 + whh[2][2]*h2 + whh[2][3]*h3;
      const float z3 = a3 + whh[3][0]*h0 + whh[3][1]*h1 + whh[3][2]*h2 + whh[3][3]*h3;
      h0 = TANHF(z0); h1 = TANHF(z1); h2 = TANHF(z2); h3 = TANHF(z3);
    }
  };

#if HAS_TDM
  const float* gbase = x + (size_t)rowBase * (T_LEN * I_DIM);
  tdm_load_tile(gbase, lds_byte_offset(buf0));
  for (int c = 0; c < NCHUNK - 1; ++c) {
    tdm_load_tile(gbase + (size_t)(c + 1) * (TCHUNK * I_DIM),
                  lds_byte_offset(((c + 1) & 1) ? buf1 : buf0));
    __builtin_amdgcn_s_wait_tensorcnt(1);                 // chunk c has landed
    asm volatile("" ::: "memory");
    compute_chunk((c & 1) ? buf1 : buf0);
    asm volatile("" ::: "memory");
  }
  __builtin_amdgcn_s_wait_tensorcnt(0);
  asm volatile("" ::: "memory");
  compute_chunk(((NCHUNK - 1) & 1) ? buf1 : buf0);
#else
  // Fallback: wave-cooperative coalesced copy (same padded LDS mapping)
  for (int c = 0; c < NCHUNK; ++c) {
    const float* gsrc = x + (size_t)rowBase * (T_LEN * I_DIM) + (size_t)c * (TCHUNK * I_DIM);
    for (int rr = 0; rr < 32; ++rr) {
      const float* gr = gsrc + (size_t)rr * (T_LEN * I_DIM);
      float* dr = buf0 + rr * ROW_DW_PAD;
      for (int k = lane; k < ROW_DW; k += 32) dr[k + (k >> 4)] = gr[k];
    }
    __syncthreads();
    compute_chunk(buf0);
    __syncthreads();
  }
#endif

  // ---- fc head: out[b,:] = h @ W_fc^T + b_fc ----
#if HAS_WMMA4
  // Stage h to LDS, then D(16x16) = A(16x4) * B(4x16) + C with C preloaded to b_fc.
  float* hstage = buf0;                                   // reuse wave-local buffer
  asm volatile("" ::: "memory");
  hstage[lane * 4 + 0] = h0;
  hstage[lane * 4 + 1] = h1;
  hstage[lane * 4 + 2] = h2;
  hstage[lane * 4 + 3] = h3;
  asm volatile("" ::: "memory");
  const int n    = lane & 15;                             // N index (cols 0..4 valid)
  const int half = lane >> 4;                             // K pair select
  const int k0 = half ? 2 : 0, k1 = half ? 3 : 1;
  v2f_ bb;
  bb.x = (n < C_DIM) ? W_fc[n * H_DIM + k0] : 0.f;        // B[k][n] = W_fc[n][k]
  bb.y = (n < C_DIM) ? W_fc[n * H_DIM + k1] : 0.f;
  const float bias = (n < C_DIM) ? b_fc[n] : 0.f;
#pragma unroll
  for (int g = 0; g < 2; ++g) {                           // rows 0-15, 16-31 of this wave
    const float* hs = hstage + g * 16 * 4;
    v2f_ aa;
    aa.x = hs[n * 4 + k0];                                // A[m][k], m = lane&15
    aa.y = hs[n * 4 + k1];
    v8f_ cc;
#pragma unroll
    for (int j = 0; j < 8; ++j) cc[j] = bias;
    v8f_ dd = __builtin_amdgcn_wmma_f32_16x16x4_f32(
        /*neg_a=*/false, aa, /*neg_b=*/false, bb,
        /*c_mod=*/(short)0, cc, /*reuse_a=*/false, /*reuse_b=*/false);
    if (n < C_DIM) {
#pragma unroll
      for (int v = 0; v < 8; ++v) {
        const int M = v + (half ? 8 : 0);
        out[(size_t)(rowBase + g * 16 + M) * C_DIM + n] = dd[v];
      }
    }
  }
#else
  const int myRow = rowBase + lane;
#pragma unroll
  for (int c2 = 0; c2 < C_DIM; ++c2) {
    out[(size_t)myRow * C_DIM + c2] = b_fc[c2]
        + W_fc[c2 * H_DIM + 0] * h0 + W_fc[c2 * H_DIM + 1] * h1
        + W_fc[c2 * H_DIM + 2] * h2 + W_fc[c2 * H_DIM + 3] * h3;
  }
#endif
}

extern "C" void kernel_launch(void* const* d_in, const int* in_sizes, int n_in,
                              void* d_out, int out_size, void* d_ws, size_t ws_size,
                              hipStream_t stream) {
  const float* x    = (const float*)d_in[0];
  const float* W_ih = (const float*)d_in[1];
  const float* W_hh = (const float*)d_in[2];
  const float* W_fc = (const float*)d_in[3];
  const float* b_fc = (const float*)d_in[4];
  float* out = (float*)d_out;

  dim3 grid(B_TOT / BLOCK_T);   // 256 workgroups
  dim3 block(BLOCK_T);          // 64 threads = 2 waves (wave32)
  rnn_fused<<<grid, block, LDS_BYTES, stream>>>(x, W_ih, W_hh, W_fc, b_fc, out);

  (void)in_sizes; (void)n_in; (void)out_size; (void)d_ws; (void)ws_size;
}